// ProbLoss_1271310320023
// MI455X (gfx1250) — compile-verified
//
#include <hip/hip_runtime.h>
#include <cstdint>

#define SIZE 128
#define BLOCK 256
#define WPB 8                        // wave32s per block
#define NBLOCKS 1024
#define TOTAL_WAVES (NBLOCKS * WPB)  // 8192 waves -> 32 rows/wave at N=262144
#define STREAM_BYTES 512             // 32 lanes * 16B: one row of one stream
#define BUF_BYTES (3 * STREAM_BYTES) // mean + logvar + target for one row
#define PIPE_DEPTH 4                 // 4-stage async ring buffer per wave
#define WAVE_LDS (PIPE_DEPTH * BUF_BYTES)  // 6 KB/wave, 48 KB/block

// ---- CDNA5 async global->LDS copy, 16B per lane (tracked by ASYNCcnt) ----
__device__ __forceinline__ void async_b128(uint32_t lds_addr, uint32_t byte_off,
                                           uint64_t sbase) {
  asm volatile("global_load_async_to_lds_b128 %0, %1, %2"
               :: "v"(lds_addr), "v"(byte_off), "s"(sbase)
               : "memory");
}
#define WAIT_ASYNC(imm) asm volatile("s_wait_asynccnt " imm ::: "memory")

// stable softplus with fast transcendentals (v_exp_f32 / v_log_f32)
__device__ __forceinline__ float softplus_fast(float x) {
  return fmaxf(x, 0.f) + __logf(1.f + __expf(-fabsf(x)));
}

// per-element contribution: diff^2/var + logvar, with /var folded into exp(-lv)
__device__ __forceinline__ float elem(float mean, float lv, float t,
                                      float ml, float mn) {
  lv = ml - softplus_fast(ml - lv);
  lv = mn + softplus_fast(lv - mn);
  float d = mean - t;
  return __fmaf_rn(d * d, __expf(-lv), lv);
}

__global__ void __launch_bounds__(BLOCK)
prob_loss_main(const float* __restrict__ inputs,   // (N, 256)
               const float* __restrict__ targets,  // (N, 128)
               const float* __restrict__ maxlv,    // (1, 128)
               const float* __restrict__ minlv,    // (1, 128)
               float* __restrict__ partials,       // (NBLOCKS,)
               int nrows) {
  __shared__ __align__(16) uint8_t smem[WPB * WAVE_LDS];
  __shared__ float wsum[WPB];

  const int lane = threadIdx.x & 31;
  const int wib  = threadIdx.x >> 5;              // wave in block
  const int gw   = blockIdx.x * WPB + wib;        // global wave id

  uint8_t* wbase = smem + wib * WAVE_LDS;
  // Flat LDS address: high 32 bits are the aperture, low 32 bits are the LDS
  // byte address that global_load_async_to_lds expects in VDST.
  const uint32_t lds_base = (uint32_t)(uintptr_t)wbase;
  const uint32_t lane16   = (uint32_t)lane * 16u;

  const uint64_t in_base = (uint64_t)(uintptr_t)inputs;
  const uint64_t tg_base = (uint64_t)(uintptr_t)targets;

  // lane i permanently owns columns 4i..4i+3
  const float4 ml = reinterpret_cast<const float4*>(maxlv)[lane];
  const float4 mn = reinterpret_cast<const float4*>(minlv)[lane];

  const int rpw  = (nrows + TOTAL_WAVES - 1) / TOTAL_WAVES;
  const int r0   = gw * rpw;
  int rend = r0 + rpw;
  if (rend > nrows) rend = nrows;
  const int cnt = (rend > r0) ? (rend - r0) : 0;

  // issue 3 async B128 copies for one row into ring slot `buf`
  auto issue = [&](int row, int buf) {
    const uint32_t lb   = lds_base + (uint32_t)buf * BUF_BYTES + lane16;
    const uint32_t moff = (uint32_t)row * 1024u + lane16;        // mean bytes
    async_b128(lb,                    moff,                       in_base);
    async_b128(lb + STREAM_BYTES,     moff + 512u,                in_base);  // logvar
    async_b128(lb + 2 * STREAM_BYTES, (uint32_t)row * 512u + lane16, tg_base);
  };

  float acc = 0.f;
  if (cnt > 0) {
    // prologue: fill up to PIPE_DEPTH-1 stages ahead
    issue(r0, 0);
    if (1 < cnt) issue(r0 + 1, 1);
    if (2 < cnt) issue(r0 + 2, 2);
    for (int i = 0; i < cnt; ++i) {
      // Keep 3 rows (9 async batches) in flight; async batches complete in
      // order, so waiting for <= 3*(rows_ahead) guarantees row i has landed.
      if (i + 3 < cnt) {
        issue(r0 + i + 3, (i + 3) & (PIPE_DEPTH - 1));
        WAIT_ASYNC("0x9");
      } else if (i + 2 < cnt) {
        WAIT_ASYNC("0x6");
      } else if (i + 1 < cnt) {
        WAIT_ASYNC("0x3");
      } else {
        WAIT_ASYNC("0x0");
      }
      const uint8_t* b = wbase + ((i & (PIPE_DEPTH - 1)) * BUF_BYTES);
      const float4 me = *reinterpret_cast<const float4*>(b + lane16);
      const float4 lv = *reinterpret_cast<const float4*>(b + STREAM_BYTES + lane16);
      const float4 tg = *reinterpret_cast<const float4*>(b + 2 * STREAM_BYTES + lane16);
      acc += elem(me.x, lv.x, tg.x, ml.x, mn.x);
      acc += elem(me.y, lv.y, tg.y, ml.y, mn.y);
      acc += elem(me.z, lv.z, tg.z, ml.z, mn.z);
      acc += elem(me.w, lv.w, tg.w, ml.w, mn.w);
    }
  }

  // deterministic wave32 reduce
  #pragma unroll
  for (int off = 16; off > 0; off >>= 1)
    acc += __shfl_down(acc, off, 32);
  if (lane == 0) wsum[wib] = acc;
  __syncthreads();
  if (threadIdx.x == 0) {
    float s = 0.f;
    #pragma unroll
    for (int w = 0; w < WPB; ++w) s += wsum[w];
    partials[blockIdx.x] = s;
  }
}

// deterministic fixed-order final reduction of per-block partials
__global__ void __launch_bounds__(256)
prob_loss_final(const float* __restrict__ partials, float* __restrict__ out, int n) {
  __shared__ float s[256];
  float v = 0.f;
  for (int i = threadIdx.x; i < n; i += 256) v += partials[i];
  s[threadIdx.x] = v;
  __syncthreads();
  #pragma unroll
  for (int st = 128; st > 0; st >>= 1) {
    if (threadIdx.x < st) s[threadIdx.x] += s[threadIdx.x + st];
    __syncthreads();
  }
  if (threadIdx.x == 0) out[0] = s[0];
}

extern "C" void kernel_launch(void* const* d_in, const int* in_sizes, int n_in,
                              void* d_out, int out_size, void* d_ws, size_t ws_size,
                              hipStream_t stream) {
  const float* inputs  = (const float*)d_in[0];  // (N, 2*SIZE) f32
  const float* targets = (const float*)d_in[1];  // (N, SIZE)   f32
  const float* maxlv   = (const float*)d_in[2];  // (1, SIZE)   f32
  const float* minlv   = (const float*)d_in[3];  // (1, SIZE)   f32
  float* out      = (float*)d_out;
  float* partials = (float*)d_ws;                // NBLOCKS floats of scratch

  const int nrows = in_sizes[1] / SIZE;          // N from targets size

  prob_loss_main<<<NBLOCKS, BLOCK, 0, stream>>>(inputs, targets, maxlv, minlv,
                                                partials, nrows);
  prob_loss_final<<<1, 256, 0, stream>>>(partials, out, NBLOCKS);
}